// CapConv2D_31533649887857
// MI455X (gfx1250) — compile-verified
//
#include <hip/hip_runtime.h>
#include <math.h>

#define CAPS_EPS 1e-4f

typedef __attribute__((ext_vector_type(2))) float v2f;
typedef __attribute__((ext_vector_type(8))) float v8f;

// weights (p,q,i,t,o) fp32, flat idx = (((p*3+q)*32+i)*16+t)*32 + o
// -> wT[o][k][i][t], k = p*3+q  (coalesced per-b slice of 4608 floats)
__global__ void caps_transpose_w(const float* __restrict__ w, float* __restrict__ wT) {
    int idx = blockIdx.x * 256 + threadIdx.x;
    if (idx >= 9 * 32 * 16 * 32) return;
    int o   = idx & 31;
    int kit = idx >> 5;            // (k*32+i)*16+t
    wT[o * 4608 + kit] = w[idx];
}

__launch_bounds__(256)
__global__ void caps_route_kernel(const float* __restrict__ x,
                                  const float* __restrict__ wsrc, int wt_flag,
                                  float* __restrict__ out,
                                  const int* __restrict__ nroutes_p)
{
    // LDS: 3200 + 4608 + 32768 floats = 162304 bytes (compiles on gfx1250)
    __shared__ __align__(16) float xlds[32 * 100];        // [a][10][10] haloed tile
    __shared__ __align__(16) float wlds[9 * 32 * 16];     // [k][a][c] (k*512 + a*16 + c)
    __shared__ __align__(16) float votes[64 * 32 * 16];   // [pix][a][c]

    const int tid  = threadIdx.x;
    const int lane = tid & 31;
    const int wv   = tid >> 5;           // 8 waves
    const int b    = blockIdx.y;         // out map 0..31
    const int tile = blockIdx.x;         // 0..63
    const int gy0  = (tile >> 3) * 8;
    const int gx0  = (tile & 7) * 8;

    // ---- Phase 1: stage x patch (zero-padded halo) and weight slice ----
    for (int e = tid; e < 3200; e += 256) {
        int i  = e / 100;
        int rr = (e % 100) / 10;
        int cc = e % 10;
        int gy = gy0 - 1 + rr;
        int gx = gx0 - 1 + cc;
        float v = 0.0f;
        if (gy >= 0 && gy < 64 && gx >= 0 && gx < 64)
            v = x[i * 4096 + gy * 64 + gx];
        xlds[e] = v;
    }
    if (wt_flag) {
        const float* wb = wsrc + b * 4608;
        for (int e = tid; e < 4608; e += 256) wlds[e] = wb[e];
    } else {
        for (int e = tid; e < 4608; e += 256) wlds[e] = wsrc[e * 32 + b];
    }
    __syncthreads();

    // ---- Phase 2: voting GEMM via V_WMMA_F32_16X16X4_F32 ----
    // Per input map a: C[64 pix x 16 atoms] = A[64 x 9(pad12)] * B[9(pad12) x 16]
    // Fragment layout (wave32): VGPR0 holds K = 2*half + 4*s, VGPR1 holds K+1.
    // K=9..11 padding handled by zeroing the A side only (0 * finite == 0);
    // all ds_loads are unconditional -> no EXEC-predicated load branches.
    {
        const int row  = lane & 15;
        const int half = lane >> 4;
        for (int ai = 0; ai < 4; ++ai) {
            const int a = wv * 4 + ai;
            const float* xa = &xlds[a * 100];
            const float* wb = &wlds[a * 16 + row];     // + k*512
            // B fragments: invariant across the 4 M-tiles -> hoisted
            v2f b0, b1, b2;
            b0.x = wb[half ? 2 * 512 : 0 * 512];       // K=0 / K=2
            b0.y = wb[half ? 3 * 512 : 1 * 512];       // K=1 / K=3
            b1.x = wb[half ? 6 * 512 : 4 * 512];       // K=4 / K=6
            b1.y = wb[half ? 7 * 512 : 5 * 512];       // K=5 / K=7
            const float b8 = wb[8 * 512];              // K=8 (finite filler for K>8)
            b2.x = b8;
            b2.y = b8;
            for (int mt = 0; mt < 4; ++mt) {
                const int m = mt * 16 + row;           // pixel index of A row
                // patch base for this pixel; k -> offset (k/3)*10 + k%3:
                // {0,1,2,10,11,12,20,21,22}
                const float* xb = xa + (m >> 3) * 10 + (m & 7);
                v2f a0, a1, a2;
                a0.x = xb[half ? 2 : 0];
                a0.y = xb[half ? 10 : 1];
                a1.x = xb[half ? 20 : 11];
                a1.y = xb[half ? 21 : 12];
                const float a8 = xb[22];
                a2.x = half ? 0.0f : a8;               // K=8 valid only in low half
                a2.y = 0.0f;                           // K=9 / K=11 -> zero
                v8f acc = {};
                acc = __builtin_amdgcn_wmma_f32_16x16x4_f32(
                    false, a0, false, b0, (short)0, acc, false, false);
                acc = __builtin_amdgcn_wmma_f32_16x16x4_f32(
                    false, a1, false, b1, (short)0, acc, false, false);
                acc = __builtin_amdgcn_wmma_f32_16x16x4_f32(
                    false, a2, false, b2, (short)0, acc, false, false);
                // C layout: vgpr r -> pixel = mt*16 + r + half*8, atom = row
                #pragma unroll
                for (int r2 = 0; r2 < 8; ++r2) {
                    const int pix = mt * 16 + r2 + half * 8;
                    votes[(pix * 32 + a) * 16 + row] = acc[r2];
                }
            }
        }
    }
    __syncthreads();

    // ---- Phase 3: dynamic routing, one pixel per wave iteration, lane = a ----
    const int nr = *nroutes_p;
    for (int pi = 0; pi < 8; ++pi) {
        const int pix = wv * 8 + pi;
        float v[16];
        {
            const float* vr = &votes[(pix * 32 + lane) * 16];
            #pragma unroll
            for (int c = 0; c < 16; ++c) v[c] = vr[c];
        }
        float lg = 0.0f;                 // logits[a] (lane-private)
        float preds[16];
        #pragma unroll 1
        for (int it = 0; it < nr; ++it) {
            // softmax over a (32 lanes)
            float mx = lg;
            #pragma unroll
            for (int s = 16; s > 0; s >>= 1) mx = fmaxf(mx, __shfl_xor(mx, s, 32));
            float ex = __expf(lg - mx);
            float sm = ex;
            #pragma unroll
            for (int s = 16; s > 0; s >>= 1) sm += __shfl_xor(sm, s, 32);
            const float r = ex / sm;
            // preds[c] = sum_a v[a][c] * r[a]  (butterfly: all lanes get full sum)
            #pragma unroll
            for (int c = 0; c < 16; ++c) preds[c] = v[c] * r;
            #pragma unroll
            for (int s = 16; s > 0; s >>= 1) {
                #pragma unroll
                for (int c = 0; c < 16; ++c) preds[c] += __shfl_xor(preds[c], s, 32);
            }
            // squash over c
            float sq = CAPS_EPS;
            #pragma unroll
            for (int c = 0; c < 16; ++c) sq += preds[c] * preds[c];
            const float nrm   = sqrtf(sq);
            const float scale = sq / ((1.0f + sq) * nrm + CAPS_EPS);
            #pragma unroll
            for (int c = 0; c < 16; ++c) preds[c] *= scale;
            // logits[a] = softmax(logits)[a] + sum_c v[a][c]*preds[c]
            float delta = 0.0f;
            #pragma unroll
            for (int c = 0; c < 16; ++c) delta += v[c] * preds[c];
            lg = r + delta;
        }
        // write preds (identical on all lanes) -> out[b][c][gy][gx]
        const int gy = gy0 + (pix >> 3);
        const int gx = gx0 + (pix & 7);
        if (lane < 16) {
            float myp = preds[0];
            #pragma unroll
            for (int c = 1; c < 16; ++c) if (lane == c) myp = preds[c];
            out[((b * 16 + lane) * 64 + gy) * 64 + gx] = myp;
        }
    }
}

extern "C" void kernel_launch(void* const* d_in, const int* in_sizes, int n_in,
                              void* d_out, int out_size, void* d_ws, size_t ws_size,
                              hipStream_t stream) {
    const float* x = (const float*)d_in[0];
    const float* w = (const float*)d_in[1];
    const int* nroutes = (const int*)d_in[2];
    float* out = (float*)d_out;

    const size_t wT_bytes = (size_t)9 * 32 * 16 * 32 * sizeof(float);
    int use_wt = (ws_size >= wT_bytes) ? 1 : 0;
    const float* wsrc = w;
    if (use_wt) {
        float* wT = (float*)d_ws;
        caps_transpose_w<<<(9 * 32 * 16 * 32 + 255) / 256, 256, 0, stream>>>(w, wT);
        wsrc = wT;
    }
    dim3 grid(64, 32);   // 64 spatial tiles x 32 out maps
    caps_route_kernel<<<grid, 256, 0, stream>>>(x, wsrc, use_wt, out, nroutes);
}